// PolyPooler_43250320671415
// MI455X (gfx1250) — compile-verified
//
#include <hip/hip_runtime.h>
#include <hip/hip_bf16.h>

// ---- problem constants (fixed by the reference setup) ----
#define PPTS   7          // points per edge
#define NPTS   14         // 2*P points per poly
#define HO_    8
#define WO_    32
#define CCH    256        // channels
#define BATCH  2
#define BASEW  256        // feat0 is 256x256 (scale 0.25 of 1024 image)

typedef __attribute__((ext_vector_type(2))) float v2f;
typedef __attribute__((ext_vector_type(8))) float v8f;

// A-matrix value: normalized poly coordinate.
// q: 0=top.x 1=top.y 2=bot.x 3=bot.y ; k: 0..7 (k==7 is zero padding)
__device__ __forceinline__ float poly_coord(const float* __restrict__ polys,
                                            int n, int k, int q) {
  int kk = (k < PPTS) ? k : (PPTS - 1);            // keep address in-bounds
  int j  = (q < 2) ? kk : (NPTS - 1 - kk);          // bottom is reversed
  float raw = polys[n * (NPTS * 2) + j * 2 + (q & 1)];
  float v = raw * (1.0f / 1024.0f);                 // exact pow2 scale
  return (k < PPTS) ? v : 0.0f;
}

// B-matrix value: 1-D linear-interp weight R[k][wo]
__device__ __forceinline__ float interp_w(int k, int wo) {
  float u = (float)(wo * (PPTS - 1)) / (float)(WO_ - 1);  // 186/31 = 6 exact at wo=31
  int i0 = (int)u; if (i0 > PPTS - 2) i0 = PPTS - 2;
  float f = u - (float)i0;
  float r = 0.0f;
  r = (k == i0)     ? (1.0f - f) : r;
  r = (k == i0 + 1) ? f          : r;
  return r;
}

// Kernel 1: topw/botw (x and y) via V_WMMA_F32_16X16X4_F32.
// One wave per 16 polys. ws layout: [topx | topy | botx | boty], each [npoly*32] f32.
__global__ __launch_bounds__(32) void poly_grid_wmma(const float* __restrict__ polys,
                                                     float* __restrict__ ws,
                                                     int npoly) {
  const int lane = threadIdx.x;
  const int h = lane >> 4;        // lane half selects K pair
  const int m = lane & 15;        // M (for A) / N (for B)
  const int n0 = blockIdx.x * 16;
  const int stride = npoly * WO_;

  #pragma unroll
  for (int q = 0; q < 4; ++q) {
    const int nrow = (n0 + m < npoly) ? (n0 + m) : (npoly - 1);
    v2f aLo, aHi;
    aLo.x = poly_coord(polys, nrow, 2 * h + 0, q);
    aLo.y = poly_coord(polys, nrow, 2 * h + 1, q);
    aHi.x = poly_coord(polys, nrow, 4 + 2 * h + 0, q);
    aHi.y = poly_coord(polys, nrow, 4 + 2 * h + 1, q);
    #pragma unroll
    for (int wt = 0; wt < 2; ++wt) {
      const int wo0 = wt * 16;
      v2f bLo, bHi;
      bLo.x = interp_w(2 * h + 0, wo0 + m);
      bLo.y = interp_w(2 * h + 1, wo0 + m);
      bHi.x = interp_w(4 + 2 * h + 0, wo0 + m);
      bHi.y = interp_w(4 + 2 * h + 1, wo0 + m);
      v8f c = {0.f, 0.f, 0.f, 0.f, 0.f, 0.f, 0.f, 0.f};
      c = __builtin_amdgcn_wmma_f32_16x16x4_f32(false, aLo, false, bLo,
                                                (short)0, c, false, false);
      c = __builtin_amdgcn_wmma_f32_16x16x4_f32(false, aHi, false, bHi,
                                                (short)0, c, false, false);
      // D: lane m holds column N=m; reg r holds row M = r + 8*h
      #pragma unroll
      for (int r = 0; r < 8; ++r) {
        int row = n0 + r + 8 * h;
        if (row < npoly)
          ws[q * stride + row * WO_ + wo0 + m] = c[r];
      }
    }
  }
}

// Kernel 2: level select + grid blend + bilinear sample over 32-channel chunk.
// block = 256 threads = one (ho,wo) position each; grid = (npoly, CCH/32).
__global__ __launch_bounds__(256) void poly_sample(
    const float* __restrict__ f0, const float* __restrict__ f1,
    const float* __restrict__ f2, const float* __restrict__ f3,
    const float* __restrict__ polys, const int* __restrict__ img_ids,
    const float* __restrict__ ws, float* __restrict__ out,
    int npoly, int use_ws) {
  const int n   = blockIdx.x;
  const int c0  = blockIdx.y * 32;
  const int tid = threadIdx.x;
  const int ho  = tid >> 5;
  const int wo  = tid & 31;

  // ---- shoelace area -> FPN level (redundant per thread, trivial) ----
  const float* pj = polys + (size_t)n * (NPTS * 2);
  float acc = 0.0f;
  #pragma unroll
  for (int j = 0; j < NPTS; ++j) {
    int jn = (j + 1) % NPTS;
    acc += pj[2 * j] * pj[2 * jn + 1] - pj[2 * jn] * pj[2 * j + 1];
  }
  float area = 0.5f * fabsf(acc);
  float s = sqrtf(area);
  float lv = floorf(4.0f + log2f(s / 224.0f + 1e-6f));
  lv = fminf(fmaxf(lv, 2.0f), 5.0f);
  const int lvl = (int)lv - 2;
  const int W   = BASEW >> lvl;
  const int HW  = W * W;
  const float* feat = (lvl == 0) ? f0 : (lvl == 1) ? f1 : (lvl == 2) ? f2 : f3;
  const int b = img_ids[n];

  // ---- top/bot resampled coords for this wo ----
  float tx, ty, bx, by;
  if (use_ws) {
    const int stride = npoly * WO_;
    tx = ws[0 * stride + n * WO_ + wo];
    ty = ws[1 * stride + n * WO_ + wo];
    bx = ws[2 * stride + n * WO_ + wo];
    by = ws[3 * stride + n * WO_ + wo];
  } else {  // inline fallback (bit-identical interp, pow2 scale commutes)
    float u = (float)(wo * (PPTS - 1)) / (float)(WO_ - 1);
    int i0 = (int)u; if (i0 > PPTS - 2) i0 = PPTS - 2;
    float f = u - (float)i0;
    int jt0 = i0, jt1 = i0 + 1;
    int jb0 = NPTS - 1 - i0, jb1 = NPTS - 2 - i0;
    tx = (pj[2 * jt0]     * (1.0f - f) + pj[2 * jt1]     * f) * (1.0f / 1024.0f);
    ty = (pj[2 * jt0 + 1] * (1.0f - f) + pj[2 * jt1 + 1] * f) * (1.0f / 1024.0f);
    bx = (pj[2 * jb0]     * (1.0f - f) + pj[2 * jb1]     * f) * (1.0f / 1024.0f);
    by = (pj[2 * jb0 + 1] * (1.0f - f) + pj[2 * jb1 + 1] * f) * (1.0f / 1024.0f);
  }

  // ---- grid point + bilinear params (reused across all channels) ----
  const float t  = (float)ho * (1.0f / 7.0f);
  const float gx = (1.0f - t) * tx + t * bx;
  const float gy = (1.0f - t) * ty + t * by;
  const float Wm1 = (float)(W - 1);
  float x = fminf(fmaxf(gx * Wm1, 0.0f), Wm1);
  float y = fminf(fmaxf(gy * Wm1, 0.0f), Wm1);
  float x0f = floorf(x), y0f = floorf(y);
  float fx = x - x0f, fy = y - y0f;
  int x0i = (int)x0f, y0i = (int)y0f;
  const int dx  = (x0i + 1 < W) ? 1 : 0;
  const int dyW = ((y0i + 1 < W) ? 1 : 0) * W;
  const float w00 = (1.0f - fx) * (1.0f - fy);
  const float w01 = fx * (1.0f - fy);
  const float w10 = (1.0f - fx) * fy;
  const float w11 = fx * fy;

  const float* base = feat + (size_t)b * CCH * HW + (size_t)c0 * HW
                           + (size_t)y0i * W + x0i;
  float* ob = out + ((size_t)n * CCH + c0) * (HO_ * WO_) + tid;

  // ---- channel loop: 4 gathers + blend; NT stores keep pyramid in L2 ----
  #pragma unroll 4
  for (int c = 0; c < 32; ++c) {
    const float* p = base + (size_t)c * HW;
    if (c + 1 < 32) __builtin_prefetch(p + HW, 0, 1);   // global_prefetch_b8
    float g00 = p[0];
    float g01 = p[dx];
    float g10 = p[dyW];
    float g11 = p[dx + dyW];
    float v = g00 * w00 + g01 * w01 + g10 * w10 + g11 * w11;
    __builtin_nontemporal_store(v, ob + (size_t)c * (HO_ * WO_));
  }
}

extern "C" void kernel_launch(void* const* d_in, const int* in_sizes, int n_in,
                              void* d_out, int out_size, void* d_ws, size_t ws_size,
                              hipStream_t stream) {
  const float* f0    = (const float*)d_in[0];
  const float* f1    = (const float*)d_in[1];
  const float* f2    = (const float*)d_in[2];
  const float* f3    = (const float*)d_in[3];
  const float* polys = (const float*)d_in[4];
  const int*   ids   = (const int*)d_in[5];
  float* out = (float*)d_out;
  float* ws  = (float*)d_ws;

  const int npoly = in_sizes[4] / (NPTS * 2);   // 512
  const size_t ws_need = (size_t)4 * npoly * WO_ * sizeof(float);  // 256 KB
  const int use_ws = (ws != nullptr && ws_size >= ws_need) ? 1 : 0;

  if (use_ws) {
    dim3 g1((npoly + 15) / 16), b1(32);
    poly_grid_wmma<<<g1, b1, 0, stream>>>(polys, ws, npoly);
  }
  dim3 g2(npoly, CCH / 32), b2(256);
  poly_sample<<<g2, b2, 0, stream>>>(f0, f1, f2, f3, polys, ids, ws, out,
                                     npoly, use_ws);
}